// MDKVMN_71305047048799
// MI455X (gfx1250) — compile-verified
//
#include <hip/hip_runtime.h>
#include <math.h>
#include <stdint.h>

// Problem constants (from reference)
#define B_    256
#define S_    500
#define MSLOT 64
#define QD    128     // Q_DIM == QA_DIM == KEY_DIM
#define VD    256     // VAL_DIM
#define NQA   20001   // qa embedding rows = 2*N_QUESTION + 1

typedef __attribute__((ext_vector_type(2))) float        v2f;
typedef __attribute__((ext_vector_type(8))) float        v8f;
typedef __attribute__((ext_vector_type(4))) unsigned int v4u;
typedef __attribute__((ext_vector_type(8))) int          v8i;
typedef __attribute__((ext_vector_type(4))) int          v4i;

// ---------------------------------------------------------------------------
// TDM descriptor group 0 + issue. Group0 bit layout (CDNA5 ISA):
//   [1:0] count=1, [31] gather_mode, [30] gather_index_size (0 = 16-bit),
//   [63:32] lds_addr, [120:64] global_addr, [127:126] type=2.
// This toolchain exposes the 6-arg builtin:
//   (v4u g0, v8i g1, v4i g2, v4i g3, v8i extra, i32 cpol)
// ---------------------------------------------------------------------------
__device__ __forceinline__ void tdm_issue(unsigned lds_off,
                                          unsigned long long gaddr,
                                          v8i g1, v4i g2, v4i g3,
                                          bool gather)
{
#if defined(__HIP_DEVICE_COMPILE__)
    v4u g0;
    g0.x = 1u | (gather ? 0x80000000u : 0u);
    g0.y = lds_off;
    g0.z = (unsigned)(gaddr & 0xFFFFFFFFull);
    g0.w = (unsigned)((gaddr >> 32) & 0x01FFFFFFull) | (2u << 30);
    const v8i extra = {};
    __builtin_amdgcn_tensor_load_to_lds(g0, g1, g2, g3, extra, 0);
#endif
}

template <short N>
__device__ __forceinline__ void tdm_wait()
{
#if defined(__HIP_DEVICE_COMPILE__)
    __builtin_amdgcn_s_wait_tensorcnt(N);
#endif
}

__device__ __forceinline__ unsigned lds_offset_of(const void* p)
{
    // Generic LDS pointer: low 32 bits are the LDS byte offset (ISA 10.2).
    return (unsigned)(uintptr_t)p;
}

// ---------------------------------------------------------------------------
// Phase A: w[B*S, 64] = softmax( gather(q_embed, q_data) @ key_mem^T )
// One block = 16 flat rows; 4 waves, each one 16x16 N-tile via 32 chained
// V_WMMA_F32_16X16X4_F32 over K = 128.
// ---------------------------------------------------------------------------
__global__ __launch_bounds__(128)
void dkvmn_attention_kernel(const int* __restrict__ q_data,
                            const float* __restrict__ q_embed_w,
                            const float* __restrict__ key_mem,
                            float* __restrict__ w_all)
{
    __shared__ float sQ[16][QD];
    __shared__ float sS[16][MSLOT];
    __shared__ int   sIdx[16];

    const int tid  = threadIdx.x;
    const int lane = tid & 31;
    const int wave = tid >> 5;          // 0..3
    const int r0   = blockIdx.x * 16;   // flat row base, r = b*S + t

    if (tid < 16) sIdx[tid] = q_data[r0 + tid];
    __syncthreads();

    for (int e = tid; e < 16 * QD; e += 128) {
        const int row = e >> 7;
        const int col = e & (QD - 1);
        sQ[row][col] = q_embed_w[(size_t)sIdx[row] * QD + col];
    }
    __syncthreads();

    const int n0   = wave * 16;
    const int rowA = lane & 15;
    const int hi   = lane >> 4;
    const int nCol = n0 + (lane & 15);

    v8f acc = {};
#pragma unroll
    for (int k0 = 0; k0 < QD; k0 += 4) {
        const int kk = k0 + 2 * hi;
        v2f a, b;
        a.x = sQ[rowA][kk];
        a.y = sQ[rowA][kk + 1];
        b.x = key_mem[nCol * QD + kk];
        b.y = key_mem[nCol * QD + kk + 1];
        acc = __builtin_amdgcn_wmma_f32_16x16x4_f32(
                  false, a, false, b, (short)0, acc, false, false);
    }

#pragma unroll
    for (int v = 0; v < 8; ++v)
        sS[v + 8 * hi][nCol] = acc[v];
    __syncthreads();

    if (tid < 16) {
        float mx = sS[tid][0];
#pragma unroll
        for (int m = 1; m < MSLOT; ++m) mx = fmaxf(mx, sS[tid][m]);
        float sum = 0.f;
#pragma unroll
        for (int m = 0; m < MSLOT; ++m) {
            const float e = __expf(sS[tid][m] - mx);
            sS[tid][m] = e;
            sum += e;
        }
        const float inv = 1.0f / sum;
#pragma unroll
        for (int m = 0; m < MSLOT; ++m) sS[tid][m] *= inv;
    }
    __syncthreads();

    for (int e = tid; e < 16 * MSLOT; e += 128)
        w_all[(size_t)r0 * MSLOT + e] = sS[e >> 6][e & (MSLOT - 1)];
}

// ---------------------------------------------------------------------------
// Phase B: recurrent scan. grid = (16 batch-groups, 16 dim-slices), 256 thr.
// Per-thread register state mv[64]. Per step, erase/add gates via WMMA
// (2 matrices x 4 K-chunks over 8 waves). The qa tile (row gather) and the
// attention tile (2D tile) for step t+1 are fetched by the Tensor Data Mover
// into double-buffered LDS while step t computes (TENSORcnt-tracked).
// ---------------------------------------------------------------------------
__global__ __launch_bounds__(256)
void dkvmn_scan_kernel(const int* __restrict__ q_data,
                       const int* __restrict__ qa_data,
                       const float* __restrict__ qa_embed_w,
                       const float* __restrict__ init_value_mem,
                       const float* __restrict__ erase_W,
                       const float* __restrict__ erase_b,
                       const float* __restrict__ add_W,
                       const float* __restrict__ add_b,
                       const float* __restrict__ w_all,
                       float* __restrict__ out)
{
    __shared__ float sWgt[2][QD][16];     // [erase|add] weight slice (16 KB)
    __shared__ float sBias[2][16];
    __shared__ float sQA[2][16][QD];      // double-buffered qa tile (16 KB)
    __shared__ float sAtt[2][16][MSLOT];  // double-buffered attention (8 KB)
    __shared__ float sPart[8][256];       // WMMA partials (8 KB)
    __shared__ float sE[256];
    __shared__ float sA[256];
    __shared__ int   sMask[2][16];

    const int bg    = blockIdx.x;
    const int dg    = blockIdx.y;
    const int tid   = threadIdx.x;
    const int lane  = tid & 31;
    const int wave  = tid >> 5;
    const int b_loc = tid >> 4;
    const int d_loc = tid & 15;
    const int b_glob = bg * 16 + b_loc;
    const int d_glob = dg * 16 + d_loc;

    // ---- constant D# group-1 descriptors -------------------------------
    // qa gather: 4-byte elems, tensor 20001 x 128, tile = 16 rows x 128
    v8i g1_qa;
    g1_qa[0] = 0x00020000;                 // data_size = 2 (4 bytes)
    g1_qa[1] = (int)(QD  << 16);           // tensor_dim0 = 128
    g1_qa[2] = (int)(NQA << 16);           // tensor_dim1 = 20001
    g1_qa[3] = (int)(QD  << 16);           // tile_dim0   = 128
    g1_qa[4] = 16;                         // tile_dim1   = 16 valid indices
    g1_qa[5] = QD;                         // tensor_dim0_stride = 128
    g1_qa[6] = 0;
    g1_qa[7] = 0;
    // attention tile: 64 x 16 tile, row stride S*64
    v8i g1_att;
    g1_att[0] = 0x00020000;
    g1_att[1] = (int)(MSLOT << 16);        // tensor_dim0 = 64
    g1_att[2] = (int)(16    << 16);        // tensor_dim1 = 16
    g1_att[3] = (int)(MSLOT << 16);        // tile_dim0   = 64
    g1_att[4] = 16;                        // tile_dim1   = 16
    g1_att[5] = S_ * MSLOT;                // tensor_dim0_stride = 32000
    g1_att[6] = 0;
    g1_att[7] = 0;
    const v4i zero4 = {};

    const unsigned long long qa_base  = (unsigned long long)(uintptr_t)qa_embed_w;
    const unsigned long long att_row0 =
        (unsigned long long)(uintptr_t)(w_all + (size_t)bg * 16 * S_ * MSLOT);

    // ---- static preloads ----------------------------------------------
    for (int e = tid; e < 2 * QD * 16; e += 256) {
        const int mat = e >> 11;
        const int rem = e & (QD * 16 - 1);
        const int k = rem >> 4, d = rem & 15;
        const float* src = mat ? add_W : erase_W;
        sWgt[mat][k][d] = src[(size_t)k * VD + dg * 16 + d];
    }
    if (tid < 32) {
        const int mat = tid >> 4, d = tid & 15;
        sBias[mat][d] = (mat ? add_b : erase_b)[dg * 16 + d];
    }

    float mv[MSLOT];
#pragma unroll
    for (int m = 0; m < MSLOT; ++m)
        mv[m] = init_value_mem[(size_t)m * VD + d_glob];

    out[((size_t)b_glob * S_ + 0) * VD + d_glob] = 0.0f;

    // ---- prologue: TDM fetch for t = 0 into buffer 0 -------------------
    if (wave == 0) {
        const int myidx =
            qa_data[(size_t)(bg * 16 + (lane & 15)) * S_ + 0];
        v4i g2, g3;
#pragma unroll
        for (int i = 0; i < 4; ++i) {
            g2[i] = (__builtin_amdgcn_readlane(myidx, 2 * i) & 0xFFFF) |
                    (__builtin_amdgcn_readlane(myidx, 2 * i + 1) << 16);
            g3[i] = (__builtin_amdgcn_readlane(myidx, 8 + 2 * i) & 0xFFFF) |
                    (__builtin_amdgcn_readlane(myidx, 9 + 2 * i) << 16);
        }
        tdm_issue(lds_offset_of(&sQA[0][0][0]), qa_base, g1_qa, g2, g3, true);
        tdm_issue(lds_offset_of(&sAtt[0][0][0]), att_row0,
                  g1_att, zero4, zero4, false);
    }
    if (wave == 1 && lane < 16)
        sMask[0][lane] = (q_data[(size_t)(bg * 16 + lane) * S_ + 0] >= 1);

    const int rowA = lane & 15;
    const int hi   = lane >> 4;
    const int nCol = lane & 15;
    const int mat  = wave >> 2;            // 0 = erase, 1 = add
    const int kc   = wave & 3;             // K-chunk

    for (int t = 0; t < S_; ++t) {
        const int cur = t & 1;
        const int nxt = cur ^ 1;
        const int tnext = t + 1;

        // -- issue TDM + mask loads for step t+1 while step t computes --
        if (tnext < S_) {
            if (wave == 0) {
                const int myidx =
                    qa_data[(size_t)(bg * 16 + (lane & 15)) * S_ + tnext];
                v4i g2, g3;
#pragma unroll
                for (int i = 0; i < 4; ++i) {
                    g2[i] = (__builtin_amdgcn_readlane(myidx, 2*i) & 0xFFFF) |
                            (__builtin_amdgcn_readlane(myidx, 2*i+1) << 16);
                    g3[i] = (__builtin_amdgcn_readlane(myidx, 8+2*i) & 0xFFFF) |
                            (__builtin_amdgcn_readlane(myidx, 9+2*i) << 16);
                }
                tdm_issue(lds_offset_of(&sQA[nxt][0][0]), qa_base,
                          g1_qa, g2, g3, true);
                tdm_issue(lds_offset_of(&sAtt[nxt][0][0]),
                          att_row0 + (unsigned long long)tnext * MSLOT * 4,
                          g1_att, zero4, zero4, false);
            }
            if (wave == 1 && lane < 16)
                sMask[nxt][lane] =
                    (q_data[(size_t)(bg * 16 + lane) * S_ + tnext] >= 1);
        }

        // -- wait for the CURRENT step's two tensor loads ----------------
        if (tnext < S_) tdm_wait<2>();
        else            tdm_wait<0>();
        __syncthreads();

        // -- erase/add gates: [16x128]@[128x16], K split over 4 chunks ---
        {
            v8f acc = {};
            const int kbase = kc * 32;
#pragma unroll
            for (int k0 = 0; k0 < 32; k0 += 4) {
                const int kk = kbase + k0 + 2 * hi;
                v2f a, b;
                a.x = sQA[cur][rowA][kk];
                a.y = sQA[cur][rowA][kk + 1];
                b.x = sWgt[mat][kk][nCol];
                b.y = sWgt[mat][kk + 1][nCol];
                acc = __builtin_amdgcn_wmma_f32_16x16x4_f32(
                          false, a, false, b, (short)0, acc, false, false);
            }
#pragma unroll
            for (int v = 0; v < 8; ++v)
                sPart[wave][v * 32 + lane] = acc[v];
        }
        __syncthreads();

        // -- reduce K-chunks, bias + sigmoid / tanh ----------------------
        {
            const int v = tid >> 5, l = tid & 31;
            const int rowM = v + 8 * (l >> 4);
            const int col  = l & 15;
            float se = sPart[0][tid] + sPart[1][tid] +
                       sPart[2][tid] + sPart[3][tid];
            float sa = sPart[4][tid] + sPart[5][tid] +
                       sPart[6][tid] + sPart[7][tid];
            se += sBias[0][col];
            sa += sBias[1][col];
            sE[rowM * 16 + col] = 1.0f / (1.0f + __expf(-se));
            sA[rowM * 16 + col] = tanhf(sa);
        }
        __syncthreads();

        // -- read from current state + masked register-state update ------
        {
            const float ev  = sE[b_loc * 16 + d_loc];
            const float av  = sA[b_loc * 16 + d_loc];
            const int   msk = sMask[cur][b_loc];

            float rd = 0.0f;
#pragma unroll
            for (int m = 0; m < MSLOT; ++m)
                rd += sAtt[cur][b_loc][m] * mv[m];

            if (tnext < S_)
                out[((size_t)b_glob * S_ + tnext) * VD + d_glob] = rd;

            if (msk) {
#pragma unroll
                for (int m = 0; m < MSLOT; ++m) {
                    const float wv = sAtt[cur][b_loc][m];
                    mv[m] = mv[m] * (1.0f - wv * ev) + wv * av;
                }
            }
        }
        __syncthreads();   // buffer `cur` free for TDM reuse next iteration
    }
}

// ---------------------------------------------------------------------------
extern "C" void kernel_launch(void* const* d_in, const int* in_sizes, int n_in,
                              void* d_out, int out_size, void* d_ws, size_t ws_size,
                              hipStream_t stream)
{
    const int*   q_data         = (const int*)  d_in[0];
    const int*   qa_data        = (const int*)  d_in[1];
    const float* q_embed_w      = (const float*)d_in[2];
    const float* qa_embed_w     = (const float*)d_in[3];
    const float* key_mem        = (const float*)d_in[4];
    const float* init_value_mem = (const float*)d_in[5];
    const float* erase_W        = (const float*)d_in[6];
    const float* erase_b        = (const float*)d_in[7];
    const float* add_W          = (const float*)d_in[8];
    const float* add_b          = (const float*)d_in[9];

    float* out   = (float*)d_out;
    float* w_all = (float*)d_ws;   // B*S*64 floats = 32.8 MB workspace

    dkvmn_attention_kernel<<<dim3((B_ * S_) / 16), dim3(128), 0, stream>>>(
        q_data, q_embed_w, key_mem, w_all);

    dkvmn_scan_kernel<<<dim3(16, 16), dim3(256), 0, stream>>>(
        q_data, qa_data, qa_embed_w, init_value_mem,
        erase_W, erase_b, add_W, add_b, w_all, out);
}